// ResCovBlock_27376121545030
// MI455X (gfx1250) — compile-verified
//
#include <hip/hip_runtime.h>
#include <hip/hip_bf16.h>
#include <math.h>

// ---------------------------------------------------------------------------
// GAT block for MI455X (gfx1250, wave32). GEMMs use V_WMMA_F32_16X16X4_F32
// to keep full fp32 precision (reference is fp32; GEMM FLOPs ~6.5G are
// negligible vs the L2-resident edge gather/scatter).
// ---------------------------------------------------------------------------

typedef float v2f __attribute__((ext_vector_type(2)));
typedef float v8f __attribute__((ext_vector_type(8)));

#define EMB   64
#define HEADS 4
#define HC    (HEADS * EMB)   // 256
#define FF    256

__device__ __forceinline__ v8f wmma4(v2f a, v2f b, v8f c) {
#if defined(__gfx1250__) && __has_builtin(__builtin_amdgcn_wmma_f32_16x16x4_f32)
    return __builtin_amdgcn_wmma_f32_16x16x4_f32(
        /*neg_a=*/false, a, /*neg_b=*/false, b,
        /*c_mod=*/(short)0, c, /*reuse_a=*/false, /*reuse_b=*/false);
#else
    asm volatile("v_wmma_f32_16x16x4_f32 %0, %1, %2, %0"
                 : "+v"(c) : "v"(a), "v"(b));
    return c;
#endif
}

// ---------------------------------------------------------------------------
// Generic fp32 WMMA GEMM: C[M,Nd] = A[M,K] @ B[K,Nd] (+bias[Nd]) (+res[M,Nd])
// One wave (32 threads) per 16x16 output tile. grid = (Nd/16, M/16).
// Two accumulators interleave independent WMMAs to hide the XDL latency of
// the D->C accumulation chain.
// ---------------------------------------------------------------------------
__global__ __launch_bounds__(32)
void wmma_gemm_f32(const float* __restrict__ A, const float* __restrict__ B,
                   const float* __restrict__ bias, const float* __restrict__ res,
                   float* __restrict__ C, int K, int Nd, int relu)
{
    const int lane = threadIdx.x & 31;
    const bool hi  = lane >= 16;
    const int  l   = lane & 15;
    const int  tn  = blockIdx.x * 16;
    const int  tm  = blockIdx.y * 16;

    // A fragment source: row (tm+l), k-pair offset 0 (lanes 0-15) or 2 (16-31)
    const float* Arow = A + (size_t)(tm + l) * K + (hi ? 2 : 0);
    // B fragment source: col (tn+l), k rows offset by 0 / 2 per lane half
    const float* Bcol = B + (size_t)(hi ? 2 : 0) * Nd + (tn + l);

    v8f acc0 = {};
    v8f acc1 = {};
    for (int k = 0; k < K; k += 8) {      // K is 64 or 256 -> multiple of 8
        v2f a0, b0, a1, b1;
        a0.x = Arow[k];     a0.y = Arow[k + 1];
        b0.x = Bcol[(size_t)k * Nd];
        b0.y = Bcol[(size_t)(k + 1) * Nd];
        a1.x = Arow[k + 4]; a1.y = Arow[k + 5];
        b1.x = Bcol[(size_t)(k + 4) * Nd];
        b1.y = Bcol[(size_t)(k + 5) * Nd];
        acc0 = wmma4(a0, b0, acc0);
        acc1 = wmma4(a1, b1, acc1);
    }
    v8f acc = acc0 + acc1;

    const int col = tn + l;
    float bv = bias ? bias[col] : 0.0f;
#pragma unroll
    for (int j = 0; j < 8; ++j) {
        int row = tm + j + (hi ? 8 : 0);
        float v = acc[j] + bv;
        if (res)  v += res[(size_t)row * Nd + col];
        if (relu) v = fmaxf(v, 0.0f);
        C[(size_t)row * Nd + col] = v;
    }
}

// ---------------------------------------------------------------------------
// Per-(node, head) attention scores: a_src/a_dst = <h[n,hh,:], att[hh,:]>
// ---------------------------------------------------------------------------
__global__ void attn_scores(const float* __restrict__ h,
                            const float* __restrict__ att_s,
                            const float* __restrict__ att_d,
                            float* __restrict__ a_s, float* __restrict__ a_d,
                            int Nn)
{
    int t = blockIdx.x * blockDim.x + threadIdx.x;
    if (t >= Nn * HEADS) return;
    int n = t >> 2, hh = t & 3;
    const float* hp = h + (size_t)n * HC + hh * EMB;
    const float* as = att_s + hh * EMB;
    const float* ad = att_d + hh * EMB;
    float s1 = 0.f, s2 = 0.f;
#pragma unroll 8
    for (int c = 0; c < EMB; ++c) {
        float v = hp[c];
        s1 += v * as[c];
        s2 += v * ad[c];
    }
    a_s[t] = s1;
    a_d[t] = s2;
}

// Monotone float<->uint map so unsigned atomicMax implements float max.
__device__ __forceinline__ unsigned flipf(float x) {
    unsigned u = __float_as_uint(x);
    return (u & 0x80000000u) ? ~u : (u | 0x80000000u);
}
__device__ __forceinline__ float unflipf(unsigned u) {
    u = (u & 0x80000000u) ? (u & 0x7FFFFFFFu) : ~u;
    return __uint_as_float(u);
}

// Pass 1: e = leaky_relu(a_src[src] + a_dst[dst]); segment-max via atomicMax.
__global__ void edge_pass1(const int* __restrict__ ei,
                           const float* __restrict__ a_s,
                           const float* __restrict__ a_d,
                           float* __restrict__ e_val,
                           unsigned* __restrict__ emax_u,
                           int E, int Nn)
{
    int e = blockIdx.x * blockDim.x + threadIdx.x;
    if (e >= E + Nn) return;
    int s = (e < E) ? ei[e]     : (e - E);   // self-loops appended
    int d = (e < E) ? ei[E + e] : (e - E);
#pragma unroll
    for (int hh = 0; hh < HEADS; ++hh) {
        float v = a_s[s * HEADS + hh] + a_d[d * HEADS + hh];
        v = (v > 0.f) ? v : 0.2f * v;
        e_val[(size_t)e * HEADS + hh] = v;
        atomicMax(&emax_u[d * HEADS + hh], flipf(v));
    }
}

// Pass 2: ex = exp(e - max); segment-sum denominator.
__global__ void edge_pass2(const int* __restrict__ ei,
                           float* __restrict__ e_val,
                           const unsigned* __restrict__ emax_u,
                           float* __restrict__ denom,
                           int E, int Nn)
{
    int e = blockIdx.x * blockDim.x + threadIdx.x;
    if (e >= E + Nn) return;
    int d = (e < E) ? ei[E + e] : (e - E);
#pragma unroll
    for (int hh = 0; hh < HEADS; ++hh) {
        float v  = e_val[(size_t)e * HEADS + hh];
        float m  = unflipf(emax_u[d * HEADS + hh]);
        float ex = __expf(v - m);
        e_val[(size_t)e * HEADS + hh] = ex;
        atomicAdd(&denom[d * HEADS + hh], ex);
    }
}

// Pass 3: gat_out[dst] += alpha * h[src]  (one thread per (edge, head)).
__global__ void edge_pass3(const int* __restrict__ ei,
                           const float* __restrict__ e_val,
                           const float* __restrict__ denom,
                           const float* __restrict__ h,
                           float* __restrict__ gat,
                           int E, int Nn)
{
    int t = blockIdx.x * blockDim.x + threadIdx.x;
    if (t >= (E + Nn) * HEADS) return;
    int e  = t >> 2;
    int hh = t & 3;
    int s = (e < E) ? ei[e]     : (e - E);
    int d = (e < E) ? ei[E + e] : (e - E);
    float alpha = e_val[(size_t)e * HEADS + hh] /
                  (denom[d * HEADS + hh] + 1e-16f);
    const float4* hp = (const float4*)(h + (size_t)s * HC + hh * EMB);
    float* op = gat + (size_t)d * HC + hh * EMB;
    __builtin_prefetch(hp, 0, 1);   // global_prefetch_b8: h tile lives in L2
#pragma unroll
    for (int i = 0; i < EMB / 4; ++i) {
        float4 hv = hp[i];
        atomicAdd(op + 4 * i + 0, alpha * hv.x);
        atomicAdd(op + 4 * i + 1, alpha * hv.y);
        atomicAdd(op + 4 * i + 2, alpha * hv.z);
        atomicAdd(op + 4 * i + 3, alpha * hv.w);
    }
}

// Whole-tensor sum / sum-of-squares for graph-mode LayerNorm.
__global__ void reduce_sums(const float* __restrict__ x, int n,
                            float* __restrict__ sum, float* __restrict__ sumsq)
{
    __shared__ float s1[256];
    __shared__ float s2[256];
    int tid = threadIdx.x;
    int i = blockIdx.x * blockDim.x + tid;
    float v = (i < n) ? x[i] : 0.f;
    s1[tid] = v;
    s2[tid] = v * v;
    __syncthreads();
    for (int off = 128; off > 0; off >>= 1) {
        if (tid < off) { s1[tid] += s1[tid + off]; s2[tid] += s2[tid + off]; }
        __syncthreads();
    }
    if (tid == 0) {
        atomicAdd(sum, s1[0]);
        atomicAdd(sumsq, s2[0]);
    }
}

// Graph LayerNorm apply: (x - mean)/(std + eps) * w[c] + b[c]
__global__ void ln_apply(const float* __restrict__ x,
                         const float* __restrict__ sum,
                         const float* __restrict__ sumsq,
                         const float* __restrict__ w,
                         const float* __restrict__ b,
                         float* __restrict__ out, int Nn)
{
    int i = blockIdx.x * blockDim.x + threadIdx.x;
    int n = Nn * EMB;
    if (i >= n) return;
    float cnt  = (float)n;
    float mean = *sum / cnt;
    float var  = fmaxf(*sumsq / cnt - mean * mean, 0.f);
    float inv  = 1.f / (sqrtf(var) + 1e-5f);
    int c = i & (EMB - 1);
    out[i] = (x[i] - mean) * inv * w[c] + b[c];
}

// ---------------------------------------------------------------------------
extern "C" void kernel_launch(void* const* d_in, const int* in_sizes, int n_in,
                              void* d_out, int out_size, void* d_ws, size_t ws_size,
                              hipStream_t stream)
{
    const float* node_attr = (const float*)d_in[0];
    const int*   ei        = (const int*)d_in[1];   // [2,E] int32 (JAX default x64-off)
    const float* Wg        = (const float*)d_in[2];
    const float* att_s     = (const float*)d_in[3];
    const float* att_d     = (const float*)d_in[4];
    const float* bias_g    = (const float*)d_in[5];
    const float* Wl        = (const float*)d_in[6];
    const float* ln1_w     = (const float*)d_in[7];
    const float* ln1_b     = (const float*)d_in[8];
    const float* W1        = (const float*)d_in[9];
    const float* b1        = (const float*)d_in[10];
    const float* W2        = (const float*)d_in[11];
    const float* b2        = (const float*)d_in[12];
    const float* ln2_w     = (const float*)d_in[13];
    const float* ln2_b     = (const float*)d_in[14];

    const int Nn = in_sizes[0] / EMB;  // 50000 (divisible by 16)
    const int E  = in_sizes[1] / 2;    // 800000

    // Workspace carve-up
    float*    h      = (float*)d_ws;                       // [Nn, 256]
    float*    gat    = h   + (size_t)Nn * HC;              // [Nn, 256]
    float*    a_s    = gat + (size_t)Nn * HC;              // [Nn, 4]
    float*    a_d    = a_s + (size_t)Nn * HEADS;           // [Nn, 4]
    unsigned* emax_u = (unsigned*)(a_d + (size_t)Nn * HEADS); // [Nn, 4]
    float*    denom  = (float*)(emax_u + (size_t)Nn * HEADS); // [Nn, 4]
    float*    e_val  = denom + (size_t)Nn * HEADS;         // [E+Nn, 4]
    float*    x1     = e_val + (size_t)(E + Nn) * HEADS;   // [Nn, 64]
    float*    xln1   = x1 + (size_t)Nn * EMB;              // [Nn, 64]
    float*    stats  = xln1 + (size_t)Nn * EMB;            // 4 scalars
    float*    ffh    = h;     // reuse: h retired after edge_pass3
    float*    x2     = gat;   // reuse: gat retired after GEMM @Wl

    // Zero accumulators (graph-capture-safe async memsets)
    hipMemsetAsync(gat,    0, (size_t)Nn * HC * sizeof(float), stream);
    hipMemsetAsync(emax_u, 0, (size_t)Nn * HEADS * sizeof(unsigned), stream); // < flip(-inf)
    hipMemsetAsync(denom,  0, (size_t)Nn * HEADS * sizeof(float), stream);
    hipMemsetAsync(stats,  0, 4 * sizeof(float), stream);

    const int BT = 256;
    dim3 wblk(32);

    // 1) h = node_attr @ Wg + bias_g            [Nn,64]x[64,256]
    wmma_gemm_f32<<<dim3(HC / 16, Nn / 16), wblk, 0, stream>>>(
        node_attr, Wg, bias_g, nullptr, h, EMB, HC, 0);

    // 2) attention scores
    attn_scores<<<(Nn * HEADS + BT - 1) / BT, BT, 0, stream>>>(
        h, att_s, att_d, a_s, a_d, Nn);

    // 3-5) edge softmax + aggregation
    int ET = E + Nn;
    edge_pass1<<<(ET + BT - 1) / BT, BT, 0, stream>>>(ei, a_s, a_d, e_val, emax_u, E, Nn);
    edge_pass2<<<(ET + BT - 1) / BT, BT, 0, stream>>>(ei, e_val, emax_u, denom, E, Nn);
    edge_pass3<<<((size_t)ET * HEADS + BT - 1) / BT, BT, 0, stream>>>(
        ei, e_val, denom, h, gat, E, Nn);

    // 6) x1 = gat @ Wl + node_attr (residual)   [Nn,256]x[256,64]
    wmma_gemm_f32<<<dim3(EMB / 16, Nn / 16), wblk, 0, stream>>>(
        gat, Wl, nullptr, node_attr, x1, HC, EMB, 0);

    // 7) graph LayerNorm 1
    int nel = Nn * EMB;
    reduce_sums<<<(nel + BT - 1) / BT, BT, 0, stream>>>(x1, nel, &stats[0], &stats[1]);
    ln_apply<<<(nel + BT - 1) / BT, BT, 0, stream>>>(
        x1, &stats[0], &stats[1], ln1_w, ln1_b, xln1, Nn);

    // 8) FFN: ffh = relu(xln1 @ W1 + b1)        [Nn,64]x[64,256]
    wmma_gemm_f32<<<dim3(FF / 16, Nn / 16), wblk, 0, stream>>>(
        xln1, W1, b1, nullptr, ffh, EMB, FF, 1);

    // 9) x2 = ffh @ W2 + b2 + xln1 (residual)   [Nn,256]x[256,64]
    wmma_gemm_f32<<<dim3(EMB / 16, Nn / 16), wblk, 0, stream>>>(
        ffh, W2, b2, xln1, x2, FF, EMB, 0);

    // 10) graph LayerNorm 2 -> d_out
    reduce_sums<<<(nel + BT - 1) / BT, BT, 0, stream>>>(x2, nel, &stats[2], &stats[3]);
    ln_apply<<<(nel + BT - 1) / BT, BT, 0, stream>>>(
        x2, &stats[2], &stats[3], ln2_w, ln2_b, (float*)d_out, Nn);
}